// AttnHGCN_30408368456115
// MI455X (gfx1250) — compile-verified
//
#include <hip/hip_runtime.h>
#include <hip/hip_bf16.h>

#define D_DIM 128

typedef __attribute__((ext_vector_type(2))) float v2f;
typedef __attribute__((ext_vector_type(8))) float v8f;

// ---- ordered-key float max helpers (memset(0) == -infinity sentinel) ----
__device__ __forceinline__ unsigned f2ord(float f) {
    unsigned u = __float_as_uint(f);
    return (u & 0x80000000u) ? ~u : (u | 0x80000000u);
}
__device__ __forceinline__ float ord2f(unsigned u) {
    return (u & 0x80000000u) ? __uint_as_float(u & 0x7fffffffu)
                             : __uint_as_float(~u);
}

// ============================================================
// P[N,128] = A[N,128] @ W[128,128] using V_WMMA_F32_16X16X4_F32
// block = 256 threads = 8 waves; each wave owns one 16-col tile
// of a 16-row tile. 32 WMMAs over K=128.
// ============================================================
__global__ __launch_bounds__(256) void proj_wmma(const float* __restrict__ A,
                                                 const float* __restrict__ W,
                                                 float* __restrict__ P, int N) {
    const int rowTile = blockIdx.x << 4;
    const int wave    = threadIdx.x >> 5;
    const int lane    = threadIdx.x & 31;
    const int colTile = wave << 4;
    const int r       = lane & 15;           // M index (A) / N index (B)
    const int koff    = (lane >> 4) << 1;    // lanes 16-31 hold K+2,K+3

    int arow = rowTile + r;
    if (arow >= N) arow = N - 1;             // clamp loads; EXEC stays all-1s
    const float* __restrict__ ap = A + (size_t)arow * D_DIM;
    const float* __restrict__ bp = W + colTile + r;

    v8f c = {};
#pragma unroll 8
    for (int k0 = 0; k0 < D_DIM; k0 += 4) {
        v2f a, b;
        a.x = ap[k0 + koff];
        a.y = ap[k0 + koff + 1];
        b.x = bp[(size_t)(k0 + koff) * D_DIM];
        b.y = bp[(size_t)(k0 + koff + 1) * D_DIM];
        c = __builtin_amdgcn_wmma_f32_16x16x4_f32(false, a, false, b,
                                                  (short)0, c, false, false);
    }
    // C/D layout: VGPR v -> lanes 0-15: M=v, lanes 16-31: M=v+8; N = lane&15
    const int mbase = rowTile + ((lane >> 4) << 3);
    const int col   = colTile + r;
    if (rowTile + 16 <= N) {
        // uniform full-tile fast path: no per-element EXEC juggling
        float* __restrict__ dst = P + (size_t)mbase * D_DIM + col;
#pragma unroll
        for (int v = 0; v < 8; ++v)
            dst[(size_t)v * D_DIM] = c[v];
    } else {
#pragma unroll
        for (int v = 0; v < 8; ++v) {
            int row = mbase + v;
            if (row < N) P[(size_t)row * D_DIM + col] = c[v];
        }
    }
}

// ============================================================
// Per-edge attention score, 1 wave per edge, float4 per lane.
// Head h covers elems [h*64, h*64+64) -> lanes 0-15 / 16-31.
// ============================================================
__global__ __launch_bounds__(256) void edge_score(
    const float* __restrict__ P, const int* __restrict__ head,
    const int* __restrict__ tail, const int* __restrict__ etype,
    const float* __restrict__ rel, float* __restrict__ score,
    unsigned* __restrict__ mkey, int E) {
    int e = (blockIdx.x << 3) + (threadIdx.x >> 5);
    if (e >= E) return;
    int lane = threadIdx.x & 31;
    int h = head[e], t = tail[e], rt = etype[e];
    float4 q  = reinterpret_cast<const float4*>(P   + (size_t)h  * D_DIM)[lane];
    float4 k  = reinterpret_cast<const float4*>(P   + (size_t)t  * D_DIM)[lane];
    float4 rv = reinterpret_cast<const float4*>(rel + (size_t)rt * D_DIM)[lane];
    float s = q.x*k.x*rv.x + q.y*k.y*rv.y + q.z*k.z*rv.z + q.w*k.w*rv.w;
    s += __shfl_xor(s, 1, 32);
    s += __shfl_xor(s, 2, 32);
    s += __shfl_xor(s, 4, 32);
    s += __shfl_xor(s, 8, 32);           // reduced within each 16-lane half
    s *= 0.125f;                         // 1/sqrt(DK=64)
    if ((lane & 15) == 0) {
        int hh = lane >> 4;
        score[((size_t)e << 1) + hh] = s;
        atomicMax(&mkey[((size_t)h << 1) + hh], f2ord(s));
    }
}

// ============================================================
// ex = exp(score - m[head]); denom[head] += ex (score overwritten)
// ============================================================
__global__ __launch_bounds__(256) void edge_expsum(
    const int* __restrict__ head, const unsigned* __restrict__ mkey,
    float* __restrict__ score, float* __restrict__ denom, int E2) {
    int idx = blockIdx.x * blockDim.x + threadIdx.x;
    if (idx >= E2) return;
    int e = idx >> 1, hh = idx & 1;
    int h = head[e];
    float m  = ord2f(mkey[((size_t)h << 1) + hh]);
    float ex = __expf(score[idx] - m);
    score[idx] = ex;
    atomicAdd(&denom[((size_t)h << 1) + hh], ex);
}

// ============================================================
// ent_agg[head] += attn * (ent_cur[tail] * rel)   (1 wave / edge)
// ============================================================
__global__ __launch_bounds__(256) void edge_aggregate(
    const float* __restrict__ entc, const int* __restrict__ head,
    const int* __restrict__ tail, const int* __restrict__ etype,
    const float* __restrict__ rel, const float* __restrict__ ex,
    const float* __restrict__ denom, float* __restrict__ agg, int E) {
    int e = (blockIdx.x << 3) + (threadIdx.x >> 5);
    if (e >= E) return;
    int lane = threadIdx.x & 31;
    int hh = lane >> 4;
    int h = head[e], t = tail[e], rt = etype[e];
    float attn = ex[((size_t)e << 1) + hh] / denom[((size_t)h << 1) + hh];
    float4 v  = reinterpret_cast<const float4*>(entc + (size_t)t  * D_DIM)[lane];
    float4 rv = reinterpret_cast<const float4*>(rel  + (size_t)rt * D_DIM)[lane];
    float* dst = agg + (size_t)h * D_DIM + (lane << 2);
    atomicAdd(dst + 0, v.x * rv.x * attn);
    atomicAdd(dst + 1, v.y * rv.y * attn);
    atomicAdd(dst + 2, v.z * rv.z * attn);
    atomicAdd(dst + 3, v.w * rv.w * attn);
}

// ============================================================
// user_agg[u] += w * ent_cur[i]     (1 wave / interaction edge)
// ============================================================
__global__ __launch_bounds__(256) void user_aggregate(
    const float* __restrict__ entc, const int* __restrict__ uidx,
    const int* __restrict__ iidx, const float* __restrict__ w,
    float* __restrict__ uagg, int E) {
    int e = (blockIdx.x << 3) + (threadIdx.x >> 5);
    if (e >= E) return;
    int lane = threadIdx.x & 31;
    int u = uidx[e], i = iidx[e];
    float ww = w[e];
    float4 v = reinterpret_cast<const float4*>(entc + (size_t)i * D_DIM)[lane];
    float* dst = uagg + (size_t)u * D_DIM + (lane << 2);
    atomicAdd(dst + 0, ww * v.x);
    atomicAdd(dst + 1, ww * v.y);
    atomicAdd(dst + 2, ww * v.z);
    atomicAdd(dst + 3, ww * v.w);
}

// ============================================================
// new = l2norm(agg); cur = new (if cur!=null); res += new
// ============================================================
__global__ __launch_bounds__(256) void norm_residual(
    const float* __restrict__ agg, float* __restrict__ cur,
    float* __restrict__ res, int N) {
    int n = (blockIdx.x << 3) + (threadIdx.x >> 5);
    if (n >= N) return;
    int lane = threadIdx.x & 31;
    size_t vidx = ((size_t)n * D_DIM >> 2) + lane;
    float4 v = reinterpret_cast<const float4*>(agg)[vidx];
    float ss = v.x*v.x + v.y*v.y + v.z*v.z + v.w*v.w;
    ss += __shfl_xor(ss, 1, 32);
    ss += __shfl_xor(ss, 2, 32);
    ss += __shfl_xor(ss, 4, 32);
    ss += __shfl_xor(ss, 8, 32);
    ss += __shfl_xor(ss, 16, 32);
    float inv = 1.0f / fmaxf(sqrtf(ss), 1e-12f);
    v.x *= inv; v.y *= inv; v.z *= inv; v.w *= inv;
    if (cur) reinterpret_cast<float4*>(cur)[vidx] = v;
    float4 r = reinterpret_cast<const float4*>(res)[vidx];
    r.x += v.x; r.y += v.y; r.z += v.z; r.w += v.w;
    reinterpret_cast<float4*>(res)[vidx] = r;
}

extern "C" void kernel_launch(void* const* d_in, const int* in_sizes, int n_in,
                              void* d_out, int out_size, void* d_ws, size_t ws_size,
                              hipStream_t stream) {
    (void)n_in; (void)out_size; (void)ws_size;
    const float* user_emb   = (const float*)d_in[0];
    const float* entity_emb = (const float*)d_in[1];
    const float* W_Q        = (const float*)d_in[2];
    const float* rel_emb    = (const float*)d_in[3];
    const int*   edge_index = (const int*)d_in[4];   // [2, E]
    const int*   edge_type  = (const int*)d_in[5];   // [E]
    const int*   inter_edge = (const int*)d_in[6];   // [2, E_UI]
    const float* inter_w    = (const float*)d_in[7]; // [E_UI]

    const int N_USR = in_sizes[0] / D_DIM;
    const int N_ENT = in_sizes[1] / D_DIM;
    const int E     = in_sizes[5];
    const int E_UI  = in_sizes[7];

    const int* head = edge_index;
    const int* tail = edge_index + E;
    const int* uidx = inter_edge;
    const int* iidx = inter_edge + E_UI;

    float* out      = (float*)d_out;
    float* user_res = out;                               // [N_USR,128]
    float* ent_res  = out + (size_t)N_USR * D_DIM;       // [N_ENT,128]

    // ---- workspace carve-out (256B aligned) ----
    char* ws = (char*)d_ws;
    size_t off = 0;
    auto carve = [&](size_t bytes) -> void* {
        void* p = ws + off;
        off += (bytes + 255) & ~(size_t)255;
        return p;
    };
    float*    ent_cur  = (float*)   carve((size_t)N_ENT * D_DIM * 4);
    float*    P        = (float*)   carve((size_t)N_ENT * D_DIM * 4); // reused as ent_agg
    float*    user_agg = (float*)   carve((size_t)N_USR * D_DIM * 4);
    float*    score    = (float*)   carve((size_t)E * 2 * 4);
    unsigned* mkey     = (unsigned*)carve((size_t)N_ENT * 2 * 4);
    float*    denom    = (float*)   carve((size_t)N_ENT * 2 * 4);

    // residual init + mutable working copy of entity embeddings
    hipMemcpyAsync(ent_cur,  entity_emb, (size_t)N_ENT * D_DIM * 4,
                   hipMemcpyDeviceToDevice, stream);
    hipMemcpyAsync(ent_res,  entity_emb, (size_t)N_ENT * D_DIM * 4,
                   hipMemcpyDeviceToDevice, stream);
    hipMemcpyAsync(user_res, user_emb,   (size_t)N_USR * D_DIM * 4,
                   hipMemcpyDeviceToDevice, stream);

    for (int hop = 0; hop < 2; ++hop) {
        hipMemsetAsync(mkey,     0, (size_t)N_ENT * 2 * 4, stream); // ord-key -inf
        hipMemsetAsync(denom,    0, (size_t)N_ENT * 2 * 4, stream);
        hipMemsetAsync(user_agg, 0, (size_t)N_USR * D_DIM * 4, stream);

        proj_wmma<<<(N_ENT + 15) / 16, 256, 0, stream>>>(ent_cur, W_Q, P, N_ENT);
        edge_score<<<(E + 7) / 8, 256, 0, stream>>>(P, head, tail, edge_type,
                                                    rel_emb, score, mkey, E);
        edge_expsum<<<(2 * E + 255) / 256, 256, 0, stream>>>(head, mkey, score,
                                                             denom, 2 * E);
        hipMemsetAsync(P, 0, (size_t)N_ENT * D_DIM * 4, stream); // P -> ent_agg
        edge_aggregate<<<(E + 7) / 8, 256, 0, stream>>>(ent_cur, head, tail,
                                                        edge_type, rel_emb,
                                                        score, denom, P, E);
        user_aggregate<<<(E_UI + 7) / 8, 256, 0, stream>>>(ent_cur, uidx, iidx,
                                                           inter_w, user_agg, E_UI);
        norm_residual<<<(N_ENT + 7) / 8, 256, 0, stream>>>(P, ent_cur, ent_res, N_ENT);
        norm_residual<<<(N_USR + 7) / 8, 256, 0, stream>>>(user_agg, nullptr,
                                                           user_res, N_USR);
    }
}